// KVPosAttentionMapping_40209483825888
// MI455X (gfx1250) — compile-verified
//
#include <hip/hip_runtime.h>

typedef float v2f __attribute__((ext_vector_type(2)));
typedef float v8f __attribute__((ext_vector_type(8)));

// Problem constants (from the reference)
constexpr int Bc = 4, Hc = 16, Nc = 1024, Mc = 1024, Dc = 64, Pc = 48;
constexpr long AW_ELEMS = (long)Bc * Hc * Nc * Mc;   // 64 Mi floats
constexpr long PT_ELEMS = (long)Nc * Mc;             // 1 Mi floats in ws
constexpr long SUMW_OFF = PT_ELEMS;                  // scalar slot in ws

// ---------------------------------------------------------------------------
// Kernel S: sum(w_pos) -> ws[SUMW_OFF]
// ---------------------------------------------------------------------------
__global__ void sumw_kernel(const float* __restrict__ w, float* __restrict__ out) {
  float s = 0.f;
  #pragma unroll
  for (int p = 0; p < Pc; ++p) s += w[p];
  out[0] = s;
}

// ---------------------------------------------------------------------------
// Kernel A: pos_term[n,m] = sum_p pos_enc[n,m,p] * w[p]
// 256 threads handle 256 consecutive (n,m) elements; the 48 KB of pos_enc for
// the block is staged through LDS with fully coalesced float4 loads, then each
// thread reduces its private 48 floats. Reads 192 MB from HBM exactly once.
// ---------------------------------------------------------------------------
__global__ __launch_bounds__(256) void pos_term_kernel(
    const float* __restrict__ pos, const float* __restrict__ w,
    float* __restrict__ pt) {
  __shared__ float s[256 * Pc];                 // 48 KB (WGP has 320 KB LDS)
  const int tid  = threadIdx.x;
  const long base = (long)blockIdx.x * 256;     // first (n,m) element
  const float4* __restrict__ src = (const float4*)pos + base * (Pc / 4);
  float4* s4 = (float4*)s;
  #pragma unroll
  for (int i = 0; i < Pc / 4; ++i)              // 12 coalesced float4 loads
    s4[tid + i * 256] = src[tid + i * 256];
  __syncthreads();
  float acc = 0.f;
  #pragma unroll
  for (int p = 0; p < Pc; ++p)
    acc += s[tid * Pc + p] * w[p];              // w[p] is wave-uniform (s_load)
  pt[base + tid] = acc;
}

// ---------------------------------------------------------------------------
// Kernel B: aw[b,n,m] = (K·K^T)[n,m] * (sumw/sqrt(H)) + pos_term[n,m] + b
// One wave -> 16x64 output strip (4 WMMA tiles along m, A-fragment reused).
// V_WMMA_F32_16X16X4_F32, K accumulated over 16 steps of 4.
// B-fragments are preloaded into a register array per m-tile so the compiler
// can clause the 16 loads and pipeline the waits instead of load->wait0->wmma.
// USE_WS=false fallback recomputes pos dot + sumw inline (no scratch needed).
// ---------------------------------------------------------------------------
template <bool USE_WS>
__global__ __launch_bounds__(256) void aw_kernel(
    const float* __restrict__ keys,
    const float* __restrict__ pt_or_pos,   // USE_WS ? pos_term : raw pos_enc
    const float* __restrict__ w_or_sumw,   // USE_WS ? &sumw    : w_pos
    const float* __restrict__ b_pos,
    float* __restrict__ aw) {
  const int lane  = threadIdx.x & 31;
  const int wave  = threadIdx.x >> 5;
  const int gt    = blockIdx.x * 8 + wave;       // 0 .. 65535
  const int mquad = gt & 15;                     // 4 m-tiles per wave
  const int ntile = (gt >> 4) & 63;
  const int batch = gt >> 10;                    // 0 .. 63 (= b*H + h)
  const int r = lane & 15;                       // row-in-tile / col-in-tile
  const int h = lane >> 4;                       // half-wave selector

  float sumw;
  if (USE_WS) {
    sumw = w_or_sumw[0];
  } else {
    sumw = 0.f;
    #pragma unroll
    for (int p = 0; p < Pc; ++p) sumw += w_or_sumw[p];
  }
  const float factor = 0.25f * sumw;             // 1/sqrt(16) * sum(w)
  const float bb = b_pos[0];

  // A fragment: rows n = ntile*16 + r ; lane half picks K pair (2h, 2h+1)
  const float* Abase = keys + ((long)batch * Nc + (long)ntile * 16 + r) * Dc + 2 * h;
  v2f a[16];
  #pragma unroll
  for (int s = 0; s < 16; ++s)
    a[s] = *(const v2f*)(Abase + s * 4);

  #pragma unroll
  for (int mt = 0; mt < 4; ++mt) {
    const int mtile = mquad * 4 + mt;
    // B fragment: B[k, m] = keys[m, k] -> same gather pattern, m-tile rows.
    // Preload all 16 K-step fragments so loads clause + waits pipeline.
    const float* Bbase = keys + ((long)batch * Mc + (long)mtile * 16 + r) * Dc + 2 * h;
    v2f b[16];
    #pragma unroll
    for (int s = 0; s < 16; ++s)
      b[s] = *(const v2f*)(Bbase + s * 4);

    v8f c = {};
    #pragma unroll
    for (int s = 0; s < 16; ++s) {
      // emits v_wmma_f32_16x16x4_f32
      c = __builtin_amdgcn_wmma_f32_16x16x4_f32(
          /*neg_a=*/false, a[s], /*neg_b=*/false, b[s],
          /*c_mod=*/(short)0, c, /*reuse_a=*/false, /*reuse_b=*/false);
    }

    // Epilogue: C layout -> VGPR i holds (n = ntile*16 + i + 8h, m = mtile*16 + r)
    const int m = mtile * 16 + r;
    float ptv[8];
    #pragma unroll
    for (int i = 0; i < 8; ++i) {
      const int  n   = ntile * 16 + i + 8 * h;
      const long idx = (long)n * Mc + m;
      if (USE_WS) {
        ptv[i] = pt_or_pos[idx];
      } else {
        const float* pp = pt_or_pos + idx * Pc;
        float acc = 0.f;
        #pragma unroll
        for (int p = 0; p < Pc; ++p) acc += pp[p] * w_or_sumw[p];
        ptv[i] = acc;
      }
    }
    #pragma unroll
    for (int i = 0; i < 8; ++i) {
      const int  n   = ntile * 16 + i + 8 * h;
      const long idx = (long)n * Mc + m;
      aw[(long)batch * Nc * Mc + idx] = c[i] * factor + ptv[i] + bb;
    }
  }
}

// ---------------------------------------------------------------------------
extern "C" void kernel_launch(void* const* d_in, const int* in_sizes, int n_in,
                              void* d_out, int out_size, void* d_ws, size_t ws_size,
                              hipStream_t stream) {
  const float* queries = (const float*)d_in[0];
  const float* keys    = (const float*)d_in[1];
  const float* values  = (const float*)d_in[2];
  const float* pos     = (const float*)d_in[3];
  const float* w_pos   = (const float*)d_in[4];
  const float* b_pos   = (const float*)d_in[5];
  float* out = (float*)d_out;
  float* ws  = (float*)d_ws;

  const bool have_ws = ws_size >= (size_t)(PT_ELEMS + 2) * sizeof(float);

  if (have_ws) {
    sumw_kernel<<<1, 1, 0, stream>>>(w_pos, ws + SUMW_OFF);
    pos_term_kernel<<<(int)(PT_ELEMS / 256), 256, 0, stream>>>(pos, w_pos, ws);
    aw_kernel<true><<<8192, 256, 0, stream>>>(keys, ws, ws + SUMW_OFF, b_pos, out);
  } else {
    aw_kernel<false><<<8192, 256, 0, stream>>>(keys, pos, w_pos, b_pos, out);
  }

  // q, k, v outputs are pure reshapes -> flat device-to-device copies
  float* q_out = out + AW_ELEMS;
  float* k_out = q_out + in_sizes[0];
  float* v_out = k_out + in_sizes[1];
  hipMemcpyAsync(q_out, queries, (size_t)in_sizes[0] * sizeof(float),
                 hipMemcpyDeviceToDevice, stream);
  hipMemcpyAsync(k_out, keys, (size_t)in_sizes[1] * sizeof(float),
                 hipMemcpyDeviceToDevice, stream);
  hipMemcpyAsync(v_out, values, (size_t)in_sizes[2] * sizeof(float),
                 hipMemcpyDeviceToDevice, stream);
}